// VertexUpdate_91096256348934
// MI455X (gfx1250) — compile-verified
//
#include <hip/hip_runtime.h>

// ---------------------------------------------------------------------------
// VertexUpdate: out = concat([b, x, b - segment_sum(c_ij, src_idx)], axis=1)
//
// Memory-bound scatter-reduce (~276 MB traffic, ~17 MFLOP => HBM roofline
// ~12us at 23.3 TB/s). Strategy:
//   K1: per-vertex init   out[3i+0]=b, out[3i+1]=x, out[3i+2]=b
//   K2: per-edge scatter  global_atomic_add_f32(out + 3*src + 2, -c_ij)
// Edge streams (indices + attrs, 256 MB read-once) use non-temporal B128
// loads + global_prefetch; the scattered f32 atomics resolve in the 192 MB
// L2 (device scope), where the 12 MB output stays resident.
// ---------------------------------------------------------------------------

typedef __attribute__((ext_vector_type(2))) unsigned long long ull2;
typedef __attribute__((ext_vector_type(4))) float f4;
typedef __attribute__((ext_vector_type(2))) float f2;

// Native CDNA5 no-return float atomic, device scope (L2 atomic units).
// No-return form => tracked with STOREcnt, no VGPR writeback.
__device__ __forceinline__ void atomic_fadd_noret(float* p, float v) {
  unsigned long long a = (unsigned long long)p;
  asm volatile("global_atomic_add_f32 %0, %1, off scope:SCOPE_DEV"
               :
               : "v"(a), "v"(v)
               : "memory");
}

__global__ __launch_bounds__(256) void edge_scatter_kernel(
    const unsigned long long* __restrict__ src_idx,  // row 0 of edgeij_pair, int64
    const float* __restrict__ eattr,                 // (n_edges, 2), c_ij = col 1
    float* __restrict__ out, int n_edges) {
  int t = blockIdx.x * blockDim.x + threadIdx.x;
  long long base = 4ll * t;
  float* out2 = out + 2;  // column of r_i partial sums

  if (base + 3 < n_edges) {
    // Prefetch the streams ~8KB ahead (speculative: OOB translation drops).
    __builtin_prefetch(src_idx + base + 1024, 0, 0);   // global_prefetch_b8
    __builtin_prefetch(eattr + 2 * base + 2048, 0, 0);

    // Non-temporal 16B loads: 4 edges per thread.
    ull2 i01 = __builtin_nontemporal_load((const ull2*)(src_idx + base));
    ull2 i23 = __builtin_nontemporal_load((const ull2*)(src_idx + base + 2));
    f4 a01 = __builtin_nontemporal_load((const f4*)(eattr + 2 * base));      // (a0,c0,a1,c1)
    f4 a23 = __builtin_nontemporal_load((const f4*)(eattr + 2 * base + 4));  // (a2,c2,a3,c3)

    atomic_fadd_noret(out2 + 3ull * (unsigned)i01.x, -a01.y);
    atomic_fadd_noret(out2 + 3ull * (unsigned)i01.y, -a01.w);
    atomic_fadd_noret(out2 + 3ull * (unsigned)i23.x, -a23.y);
    atomic_fadd_noret(out2 + 3ull * (unsigned)i23.y, -a23.w);
  } else {
    // Tail (n_edges not a multiple of 4).
    for (long long e = base; e < n_edges; ++e) {
      unsigned idx = (unsigned)src_idx[e];
      float c = eattr[2 * e + 1];
      atomic_fadd_noret(out2 + 3ull * idx, -c);
    }
  }
}

__global__ __launch_bounds__(256) void vertex_init_kernel(
    const float* __restrict__ vattr, float* __restrict__ out, int n_vertices) {
  int i = blockIdx.x * blockDim.x + threadIdx.x;
  if (i < n_vertices) {
    f2 v = *(const f2*)(vattr + 2ull * i);   // global_load_b64: (b_i, x_i)
    out[3ull * i + 0] = v.x;                 // b_i
    out[3ull * i + 1] = v.y;                 // x_i
    out[3ull * i + 2] = v.x;                 // r_i starts at b_i; edges subtract c_ij
  }
}

extern "C" void kernel_launch(void* const* d_in, const int* in_sizes, int n_in,
                              void* d_out, int out_size, void* d_ws, size_t ws_size,
                              hipStream_t stream) {
  // Inputs (setup_inputs order):
  //   d_in[0] vertex_attr  f32  (N_V, 2)
  //   d_in[1] edgeij_pair  i64  (2, N_E)   -> row 0 = source indices
  //   d_in[2] edge_attr    f32  (N_E, 2)   -> col 1 = c_ij
  //   d_in[3] g, d_in[4] batch (unused by the reference output)
  const float* vattr = (const float*)d_in[0];
  const unsigned long long* src_idx = (const unsigned long long*)d_in[1];
  const float* eattr = (const float*)d_in[2];
  float* out = (float*)d_out;

  const int n_vertices = in_sizes[0] / 2;
  const int n_edges    = in_sizes[2] / 2;

  // K1: fully initializes all out_size = 3*n_vertices floats (harness poisons d_out).
  vertex_init_kernel<<<(n_vertices + 255) / 256, 256, 0, stream>>>(vattr, out, n_vertices);

  // K2: 4 edges per thread; stream-ordered after K1 so init precedes atomics.
  const int n_threads = (n_edges + 3) / 4;
  edge_scatter_kernel<<<(n_threads + 255) / 256, 256, 0, stream>>>(src_idx, eattr, out, n_edges);
}